// Reduce_87230785782268
// MI455X (gfx1250) — compile-verified
//
#include <hip/hip_runtime.h>
#include <hip/hip_bf16.h>
#include <math.h>

// ---------------------------------------------------------------------------
// Pyramidal conv encoder for MI455X (gfx1250), fp32 via V_WMMA_F32_16X16X4_F32.
//   depth loop (12 levels, T: 4096 -> 1):
//     conv1 (stride2, 257->512ch, K=5)  -> l, relu+mask -> r   [WMMA GEMM]
//     conv2 (stride1, 256->256ch, K=5) on r; h = relu(l + r)   [WMMA GEMM]
//     bookkeeping: N=(N-1)/2+1, capture h[:,:,0] on finish
//   final: stable sort by finish depth.
// LDS tiles are [position][channel] (stride 66) so B fragments are single
// ds_load_b64; weights are pair-interleaved so A fragments are single
// global_load_b64 (lane-coalesced over output channels, L2-resident).
// ---------------------------------------------------------------------------

typedef __attribute__((ext_vector_type(2))) float v2f;
typedef __attribute__((ext_vector_type(8))) float v8f;

#define NB 64     // batch
#define NH 256    // hidden channels
#define KS 5
#define T0 4096
#define NDEPTH 12
#define CSTR 66   // LDS row stride (floats): 64 channels + 2 pad

__device__ inline v8f wmma_f32_4(v2f a, v2f b, v8f c) {
  // D = A(16x4,f32) * B(4x16,f32) + C(16x16,f32)
  return __builtin_amdgcn_wmma_f32_16x16x4_f32(
      /*neg_a=*/false, a, /*neg_b=*/false, b,
      /*c_mod=*/(short)0, c, /*reuse_a=*/false, /*reuse_b=*/false);
}

// Repack w1 (512, 257, 5) -> w1q[k][c/2][o][c&1], c in [0,256) (depth ch peeled).
// Pair-interleave channels so an A fragment is one aligned b64 load.
__global__ void repack_w1_kernel(const float* __restrict__ w1, float* __restrict__ w1q) {
  int idx = blockIdx.x * 256 + threadIdx.x;            // 5*256*512 elements
  if (idx >= 5 * 256 * 512) return;
  int r  = idx & 1;
  int o  = (idx >> 1) & 511;
  int cp = (idx >> 10) & 127;
  int k  = idx >> 17;
  int c  = cp * 2 + r;
  w1q[idx] = w1[(size_t)o * (257 * KS) + c * KS + k];
}

// Repack w2 (256, 256, 5) -> w2q[k][c/2][o][c&1].
__global__ void repack_w2_kernel(const float* __restrict__ w2, float* __restrict__ w2q) {
  int idx = blockIdx.x * 256 + threadIdx.x;            // 5*256*256 elements
  if (idx >= 5 * 256 * 256) return;
  int r  = idx & 1;
  int o  = (idx >> 1) & 255;
  int cp = (idx >> 9) & 127;
  int k  = idx >> 16;
  int c  = cp * 2 + r;
  w2q[idx] = w2[((size_t)o * 256 + c) * KS + k];
}

__global__ void init_state_kernel(const int* __restrict__ Nin, int* __restrict__ Ncur,
                                  int* __restrict__ done) {
  int b = threadIdx.x;
  if (b < NB) { Ncur[b] = Nin[b]; done[b] = 0; }
}

// ---------------------------------------------------------------------------
// conv1: stride-2, cin=256(+depth ch in epilogue), cout=512.
// Block: 256 threads = 8 waves. Each wave: 1 M-tile (16 out ch) x 4 pos-tiles
// (64 positions, A-fragment register reuse x4).
// grid = (ceil(Tout/64), B, 4 m-splits of 128 ch).
// Activations staged in LDS in 4 chunks of 64 channels (input mask applied).
// ---------------------------------------------------------------------------
__global__ __launch_bounds__(256)
void conv1_kernel(const float* __restrict__ h, const float* __restrict__ w1,
                  const float* __restrict__ w1q, const float* __restrict__ b1,
                  const int* __restrict__ Ncur,
                  float* __restrict__ lbuf, float* __restrict__ rbuf,
                  int Tin, int Tout, float dval) {
  __shared__ float smem[131 * CSTR];                   // [pos 0..130][ch 0..63]
  const int b    = blockIdx.y;
  const int t0   = blockIdx.x * 64;
  const int wave = threadIdx.x >> 5;
  const int mbase = blockIdx.z * 128 + wave * 16;      // out-channel base (0..496)
  const int lane = threadIdx.x & 31;
  const int l15  = lane & 15;
  const int half = lane >> 4;
  const int kb   = half * 2;
  const int nb   = Ncur[b];
  const int nnext = (nb - 1) / 2 + 1;
  const int pin0 = 2 * t0 - 2;                         // input pos of tile col 0
  const float* hb = h + (size_t)b * NH * Tin;

  v8f acc[4] = {};

  for (int ch = 0; ch < 4; ++ch) {
    const int cbase = ch * 64;
    __syncthreads();
    for (int i = threadIdx.x; i < 64 * 131; i += 256) {
      int cr = i / 131;                                // channel (coalesced along p)
      int p  = i - cr * 131;                           // position column
      int pin = pin0 + p;
      float v = 0.0f;
      if (pin >= 0 && pin < nb) {
        v = hb[(size_t)(cbase + cr) * Tin + pin];      // seq-mask folded into load
        if (ch < 3)                                    // prefetch next chunk
          __builtin_prefetch(&hb[(size_t)(cbase + 64 + cr) * Tin + pin], 0, 0);
      }
      smem[p * CSTR + cr] = v;                         // transposed store
    }
    __syncthreads();

    for (int k = 0; k < KS; ++k) {
      // A fragment source: pair-interleaved weights, lane-coalesced over o.
      const float* wsl = w1q + (((size_t)k * 128) * 512 + (size_t)(cbase / 2) * 512
                                + mbase + l15) * 2;
      for (int c0 = 0; c0 < 64; c0 += 4) {
        v2f afr = *(const v2f*)(wsl + (size_t)((c0 + kb) >> 1) * 1024);
#pragma unroll
        for (int pt = 0; pt < 4; ++pt) {
          int col = 2 * (pt * 16 + l15) + k;           // stride-2 gather
          v2f bfr = *(const v2f*)(smem + col * CSTR + c0 + kb);  // one ds_load_b64
          acc[pt] = wmma_f32_4(afr, bfr, acc[pt]);
        }
      }
    }
  }

  // Epilogue: bias + analytic depth-channel term; split into l / relu+mask r.
#pragma unroll
  for (int pt = 0; pt < 4; ++pt) {
    int t = t0 + pt * 16 + l15;
    if (t >= Tout) continue;
    float s[KS];
#pragma unroll
    for (int k = 0; k < KS; ++k) {
      int pin = 2 * t - 2 + k;
      s[k] = (pin >= 0 && pin < nb) ? 1.0f : 0.0f;
    }
#pragma unroll
    for (int i = 0; i < 8; ++i) {
      int o = mbase + half * 8 + i;                    // C/D: M = i + 8*half
      float dterm = 0.0f;
#pragma unroll
      for (int k = 0; k < KS; ++k)
        dterm += w1[(size_t)o * (257 * KS) + 256 * KS + k] * s[k];
      float val = acc[pt][i] + b1[o] + dterm * dval;
      if (o < NH) {
        lbuf[((size_t)b * NH + o) * Tout + t] = val;
      } else {
        val = fmaxf(val, 0.0f);
        if (t >= nnext) val = 0.0f;                    // seq-mask with updated N
        rbuf[((size_t)b * NH + (o - NH)) * Tout + t] = val;
      }
    }
  }
}

// ---------------------------------------------------------------------------
// conv2: stride-1, 256->256; h_new = relu(l + conv2(r) + b2), written over l.
// grid = (ceil(Tout/64), B, 2 m-splits of 128 ch).
// ---------------------------------------------------------------------------
__global__ __launch_bounds__(256)
void conv2_kernel(const float* __restrict__ rbuf, const float* __restrict__ w2q,
                  const float* __restrict__ b2, float* __restrict__ lh, int Tout) {
  __shared__ float smem[68 * CSTR];                    // [pos 0..67][ch 0..63]
  const int b    = blockIdx.y;
  const int t0   = blockIdx.x * 64;
  const int wave = threadIdx.x >> 5;
  const int mbase = blockIdx.z * 128 + wave * 16;
  const int lane = threadIdx.x & 31;
  const int l15  = lane & 15;
  const int half = lane >> 4;
  const int kb   = half * 2;
  const int pin0 = t0 - 2;
  const float* rb = rbuf + (size_t)b * NH * Tout;

  v8f acc[4] = {};

  for (int ch = 0; ch < 4; ++ch) {
    const int cbase = ch * 64;
    __syncthreads();
    for (int i = threadIdx.x; i < 64 * 68; i += 256) {
      int cr = i / 68;
      int p  = i - cr * 68;
      int pin = pin0 + p;
      float v = 0.0f;
      if (pin >= 0 && pin < Tout) v = rb[(size_t)(cbase + cr) * Tout + pin];
      smem[p * CSTR + cr] = v;
    }
    __syncthreads();

    for (int k = 0; k < KS; ++k) {
      const float* wsl = w2q + (((size_t)k * 128) * 256 + (size_t)(cbase / 2) * 256
                                + mbase + l15) * 2;
      for (int c0 = 0; c0 < 64; c0 += 4) {
        v2f afr = *(const v2f*)(wsl + (size_t)((c0 + kb) >> 1) * 512);
#pragma unroll
        for (int pt = 0; pt < 4; ++pt) {
          int col = pt * 16 + l15 + k;
          v2f bfr = *(const v2f*)(smem + col * CSTR + c0 + kb);
          acc[pt] = wmma_f32_4(afr, bfr, acc[pt]);
        }
      }
    }
  }

#pragma unroll
  for (int pt = 0; pt < 4; ++pt) {
    int t = t0 + pt * 16 + l15;
    if (t >= Tout) continue;
#pragma unroll
    for (int i = 0; i < 8; ++i) {
      int o = mbase + half * 8 + i;
      size_t idx = ((size_t)b * NH + o) * Tout + t;
      float val = acc[pt][i] + b2[o] + lh[idx];
      lh[idx] = fmaxf(val, 0.0f);                      // h_new in place over l
    }
  }
}

// Per-depth bookkeeping: capture h[:, :, 0] when N first reaches <= 1.
__global__ void capture_kernel(const float* __restrict__ hnew, int* __restrict__ Ncur,
                               int* __restrict__ done, int* __restrict__ finish,
                               float* __restrict__ captured, int Tout, int depth) {
  int b = blockIdx.x;
  int c = threadIdx.x;
  int nb = Ncur[b];
  int nnext = (nb - 1) / 2 + 1;
  bool newly = (done[b] == 0) && (nnext <= 1);
  __syncthreads();                                     // all reads of done[] first
  if (newly) captured[b * NH + c] = hnew[((size_t)b * NH + c) * Tout];
  if (c == 0) {
    if (newly) { finish[b] = depth; done[b] = 1; }
    Ncur[b] = nnext;
  }
}

// Stable rank sort by finish depth (B=64, O(B^2) in one block).
__global__ void sort_kernel(const float* __restrict__ captured,
                            const int* __restrict__ finish, float* __restrict__ out) {
  int b = threadIdx.x;                                 // 64 threads
  int fb = finish[b];
  int rank = 0;
  for (int j = 0; j < NB; ++j) {
    int fj = finish[j];
    if (fj < fb || (fj == fb && j < b)) rank++;
  }
  for (int c = 0; c < NH; ++c)
    out[(size_t)rank * NH + c] = captured[b * NH + c];
}

extern "C" void kernel_launch(void* const* d_in, const int* in_sizes, int n_in,
                              void* d_out, int out_size, void* d_ws, size_t ws_size,
                              hipStream_t stream) {
  const float* h0  = (const float*)d_in[0];
  const int*   Nin = (const int*)d_in[1];
  const float* w1  = (const float*)d_in[2];
  const float* b1  = (const float*)d_in[3];
  const float* w2  = (const float*)d_in[4];
  const float* b2  = (const float*)d_in[5];
  float* out = (float*)d_out;

  // Workspace carving: [w1q 2.5MB][w2q 1.25MB] ... 4MB: [lA 128MB][r 128MB][lB 64MB][state]
  float* w1q = (float*)d_ws;
  float* w2q = w1q + (size_t)5 * 256 * 512;
  char*  base = (char*)d_ws + (4u << 20);
  float* lA = (float*)base;
  float* rb = lA + (size_t)NB * NH * 2048;
  float* lB = rb + (size_t)NB * NH * 2048;
  int*   Ncur = (int*)(lB + (size_t)NB * NH * 1024);
  int*   done = Ncur + NB;
  int*   finish = done + NB;
  float* captured = (float*)(finish + NB);

  repack_w1_kernel<<<(5 * 256 * 512 + 255) / 256, 256, 0, stream>>>(w1, w1q);
  repack_w2_kernel<<<(5 * 256 * 256 + 255) / 256, 256, 0, stream>>>(w2, w2q);
  init_state_kernel<<<1, 64, 0, stream>>>(Nin, Ncur, done);

  const float* hin = h0;
  int Tin = T0;
  int ping = 0;
  for (int depth = 0; depth < NDEPTH; ++depth) {
    int Tout = Tin >> 1;
    float* lbuf = ping ? lB : lA;
    float dval = log1pf((float)depth);
    dim3 g1((Tout + 63) / 64, NB, 4);
    conv1_kernel<<<g1, 256, 0, stream>>>(hin, w1, w1q, b1, Ncur, lbuf, rb, Tin, Tout, dval);
    dim3 g2((Tout + 63) / 64, NB, 2);
    conv2_kernel<<<g2, 256, 0, stream>>>(rb, w2q, b2, lbuf, Tout);
    capture_kernel<<<NB, NH, 0, stream>>>(lbuf, Ncur, done, finish, captured, Tout, depth);
    hin = lbuf;
    ping ^= 1;
    Tin = Tout;
  }
  sort_kernel<<<1, 64, 0, stream>>>(captured, finish, out);
}